// Leg_fit_21595095564537
// MI455X (gfx1250) — compile-verified
//
#include <hip/hip_runtime.h>
#include <math.h>

#define T_LEN 1000
#define N_NRN 400
#define N_EXC 320
#define N_PRE 8

typedef __attribute__((ext_vector_type(2))) float v2f;
typedef __attribute__((ext_vector_type(8))) float v8f;

// ---------------------------------------------------------------------------
// Kernel 1: dopa[t] = sum_k R_K[k] * z_pad[t+k]; then reverse exponential scan
//           g[s] = dopa[s] + DECAY_E * g[s+1]   (g folds the etrace recurrence)
// ---------------------------------------------------------------------------
__global__ void dopa_g_kernel(const float* __restrict__ z,
                              const float* __restrict__ z_cn_pre,
                              const int* __restrict__ idx_cn_p,
                              float* __restrict__ g) {
    __shared__ float zpad[T_LEN + N_PRE - 1];   // 1007
    __shared__ float dopa[T_LEN];
    const int idx = idx_cn_p[0];

    // z_pad = concat(z_cn_pre[0,:,0], z[:, :, idx])[1:]
    for (int m = threadIdx.x; m < T_LEN + N_PRE - 1; m += blockDim.x) {
        zpad[m] = (m < N_PRE - 1) ? z_cn_pre[m + 1]
                                  : z[(m - (N_PRE - 1)) * N_NRN + idx];
    }
    __syncthreads();

    // R_K computed in double, cast to float (matches numpy reference)
    float rk[N_PRE];
#pragma unroll
    for (int k = 0; k < N_PRE; ++k) {
        double tt = (double)(N_PRE - k);
        rk[k] = (float)(1.379 * tt / 200.0 * exp(1.0 - tt / 200.0)
                      - 0.27  * tt / 1023.0 * exp(1.0 - tt / 1023.0));
    }

    for (int t = threadIdx.x; t < T_LEN; t += blockDim.x) {
        float s = 0.f;
#pragma unroll
        for (int k = 0; k < N_PRE; ++k) s += rk[k] * zpad[t + k];
        dopa[t] = s;
    }
    __syncthreads();

    if (threadIdx.x == 0) {
        const float de = (float)exp(-1.0 / 400.0);   // DECAY_E
        float acc = 0.f;
        for (int s = T_LEN - 1; s >= 0; --s) {
            acc = dopa[s] + de * acc;
            g[s] = acc;
        }
    }
}

// ---------------------------------------------------------------------------
// Kernel 2: per-neuron exponential filters; emit GEMM operands transposed
//   A1t[n,t] = g[t]*z_prev[t,n]   B1t[n,t] = sp[t,n]
//   A2t[n,t] = g[t]*sm[t,n]       B2t[n,t] = z[t,n]
// stored row-major [N_NRN, T_LEN] so the WMMA kernel does contiguous b64 loads
// ---------------------------------------------------------------------------
__global__ void scan_kernel(const float* __restrict__ z,
                            const float* __restrict__ g,
                            float* __restrict__ A1t, float* __restrict__ B1t,
                            float* __restrict__ A2t, float* __restrict__ B2t) {
    __shared__ float gs[T_LEN];
    for (int i = threadIdx.x; i < T_LEN; i += blockDim.x) gs[i] = g[i];
    __syncthreads();

    const int n = blockIdx.x * blockDim.x + threadIdx.x;
    if (n >= N_NRN) return;

    const float dp = (float)exp(-1.0 / 30.0);   // DECAY_P
    const float dm = (float)exp(-1.0 / 30.0);   // DECAY_M
    const float aP = 0.01f * 0.5f;              // A_P = 0.005
    const float aM = 1.05f * 0.01f * 0.5f;      // A_M = 0.00525

    float sp = 0.f, sm = 0.f, zprev = 0.f;
    float* a1 = A1t + (size_t)n * T_LEN;
    float* b1 = B1t + (size_t)n * T_LEN;
    float* a2 = A2t + (size_t)n * T_LEN;
    float* b2 = B2t + (size_t)n * T_LEN;

    for (int t = 0; t < T_LEN; ++t) {
        float zt = z[t * N_NRN + n];
        sp = sp * dp + aP * zt;        // exp_conv of z
        sm = sm * dm - aM * zprev;     // exp_conv of z_prev with gain -A_M
        float gt = gs[t];
        a1[t] = gt * zprev;
        b1[t] = sp;
        a2[t] = gt * sm;
        b2[t] = zt;
        zprev = zt;
    }
}

// ---------------------------------------------------------------------------
// Kernel 3: dw = γ·(A1ᵀB1 + A2ᵀB2) on the f32 WMMA pipe; one wave per 16x16
// tile. A (16x4 f32) lane layout: lanes 0-15 -> M=lane, K={0,1}; lanes 16-31
// -> M=lane-16, K={2,3}. B (4x16) mirrors it. Both reduce to a single
// contiguous float2 load per operand per K-step from the [N,T] layout.
// ---------------------------------------------------------------------------
__global__ void __launch_bounds__(32)
gemm_wmma_kernel(const float* __restrict__ A1t, const float* __restrict__ B1t,
                 const float* __restrict__ A2t, const float* __restrict__ B2t,
                 const float* __restrict__ w_rec_ee, float* __restrict__ out) {
    const int jb = blockIdx.x * 16;       // output col tile
    const int ib = blockIdx.y * 16;       // output row tile
    const int lane = threadIdx.x;
    const int half = lane >> 4;           // 0: K 0..1, 1: K 2..3
    const int lm = lane & 15;

    const float* pa1 = A1t + (size_t)(ib + lm) * T_LEN + 2 * half;
    const float* pb1 = B1t + (size_t)(jb + lm) * T_LEN + 2 * half;
    const float* pa2 = A2t + (size_t)(ib + lm) * T_LEN + 2 * half;
    const float* pb2 = B2t + (size_t)(jb + lm) * T_LEN + 2 * half;

    v8f acc = {};
#pragma unroll 2
    for (int kb = 0; kb < T_LEN; kb += 4) {
        v2f a1 = *(const v2f*)(pa1 + kb);
        v2f b1 = *(const v2f*)(pb1 + kb);
        v2f a2 = *(const v2f*)(pa2 + kb);
        v2f b2 = *(const v2f*)(pb2 + kb);
        acc = __builtin_amdgcn_wmma_f32_16x16x4_f32(
                  false, a1, false, b1, (short)0, acc, false, false);
        acc = __builtin_amdgcn_wmma_f32_16x16x4_f32(
                  false, a2, false, b2, (short)0, acc, false, false);
    }

    const float eg = 0.01f;  // ETRACE_GAIN
#pragma unroll
    for (int v = 0; v < 8; ++v) {
        int row = ib + v + 8 * half;      // C/D layout: VGPR v, lane-half
        int col = jb + lm;
        float wv = w_rec_ee[row * N_EXC + col];
        float dwv = eg * acc[v];
        if (wv > 0.5f) dwv = fminf(0.f, dwv);   // w > WMAX  -> min(0, dw)
        if (wv < 0.f)  dwv = fmaxf(0.f, dwv);   // w < 0     -> max(0, dw)
        out[row * N_EXC + col] = -dwv;
    }
}

// ---------------------------------------------------------------------------
extern "C" void kernel_launch(void* const* d_in, const int* in_sizes, int n_in,
                              void* d_out, int out_size, void* d_ws, size_t ws_size,
                              hipStream_t stream) {
    const float* z         = (const float*)d_in[0];   // [1,1000,400]
    const float* z_cn_pre  = (const float*)d_in[1];   // [1,8,1]
    const float* w_rec_ee  = (const float*)d_in[2];   // [320,320]
    const int*   idx_cn    = (const int*)d_in[3];     // scalar
    float* out = (float*)d_out;                       // [320,320]

    char* ws = (char*)d_ws;
    const size_t matBytes = (size_t)N_NRN * T_LEN * sizeof(float);  // 1.6 MB
    float* g   = (float*)(ws);
    float* A1t = (float*)(ws + 4096);
    float* B1t = (float*)(ws + 4096 + matBytes);
    float* A2t = (float*)(ws + 4096 + 2 * matBytes);
    float* B2t = (float*)(ws + 4096 + 3 * matBytes);

    dopa_g_kernel<<<1, 256, 0, stream>>>(z, z_cn_pre, idx_cn, g);

    scan_kernel<<<(N_NRN + 127) / 128, 128, 0, stream>>>(z, g, A1t, B1t, A2t, B2t);

    dim3 grid(N_EXC / 16, N_EXC / 16);   // 20 x 20 tiles
    gemm_wmma_kernel<<<grid, 32, 0, stream>>>(A1t, B1t, A2t, B2t, w_rec_ee, out);
}